// SubjectLayers_12386685682102
// MI455X (gfx1250) — compile-verified
//
#include <hip/hip_runtime.h>

// out[b] = x[b] @ W[sid[b]] + bias[sid[b]]
// B=8192, IN=2048, OUT=3072, S=8, fp32.
// Strategy: deterministic stable counting-sort of rows by subject, then a
// grouped GEMM over 64-row same-subject tiles using V_WMMA_F32_16X16X4_F32,
// with a 2-stage (double-buffered LDS + register-staged) K pipeline.

typedef float v2f __attribute__((ext_vector_type(2)));
typedef float v8f __attribute__((ext_vector_type(8)));

#define NSUBJ 8
#define BM 64     // rows per block tile (per-subject)
#define BN 128    // cols per block tile
#define BK 32     // K slab per stage
#define KPAD 4    // row stride 36 floats = 144B: 16B multiple, conflict-free b64 reads

// ---------------------------------------------------------------------------
// Phase A: stable counting sort of row indices by subject + tile descriptors.
// Single block; deterministic (stable order, exact per-chunk counts).
// tiles[i] = {subject, startIndexInPerm, validRows (0..BM), 0}
// ---------------------------------------------------------------------------
__global__ __launch_bounds__(256)
void sl_build_perm(const int* __restrict__ sid, int* __restrict__ perm,
                   int4* __restrict__ tiles, int B, int numTiles) {
    __shared__ int cnt[NSUBJ], off[NSUBJ], run[NSUBJ], chunkCnt[NSUBJ];
    __shared__ int sids[256];
    const int t = threadIdx.x;

    if (t < NSUBJ) cnt[t] = 0;
    __syncthreads();
    for (int r = t; r < B; r += 256) atomicAdd(&cnt[sid[r]], 1);
    __syncthreads();
    if (t == 0) {
        int a = 0;
        for (int s = 0; s < NSUBJ; s++) { off[s] = a; a += cnt[s]; run[s] = 0; }
    }
    __syncthreads();

    const int nchunks = B / 256;   // B is a multiple of 256 (8192)
    for (int c = 0; c < nchunks; c++) {
        const int r  = c * 256 + t;
        const int ms = sid[r];
        sids[t] = ms;
        __syncthreads();
        int rank = 0;
        for (int j = 0; j < t; j++) rank += (sids[j] == ms) ? 1 : 0;
        perm[off[ms] + run[ms] + rank] = r;
        if (t < NSUBJ) {
            int cc = 0;
            for (int j = 0; j < 256; j++) cc += (sids[j] == t) ? 1 : 0;
            chunkCnt[t] = cc;
        }
        __syncthreads();
        if (t < NSUBJ) run[t] += chunkCnt[t];
        __syncthreads();
    }

    if (t == 0) {
        int tix = 0;
        for (int s = 0; s < NSUBJ; s++) {
            const int c0 = off[s], n = cnt[s];
            for (int p = 0; p < n; p += BM)
                tiles[tix++] = make_int4(s, c0 + p, min(BM, n - p), 0);
        }
        for (; tix < numTiles; tix++) tiles[tix] = make_int4(0, 0, 0, 0);
    }
}

// ---------------------------------------------------------------------------
// Phase B: grouped GEMM. Block = 256 threads (8 waves = 2M x 4N).
// Block tile 64x128, K-step 32. Each wave owns 32x32 via 4 f32 WMMA
// accumulators: 4 wmma per 4 ds_load_b64 (a0,a1,b0,b1 fragments).
//
// 2-stage pipeline: while the WMMA loop consumes LDS buffer p, next slab's
// global loads are in flight into registers; after the compute they are
// written to buffer p^1. One barrier per K slab.
//
// WMMA f32 16x16x4 fragment layout (ISA 7.12.2, 32-bit A 16x4):
//   lanes 0-15 : M=lane,    v.x=K+0, v.y=K+1
//   lanes 16-31: M=lane-16, v.x=K+2, v.y=K+3
// B 4x16 symmetric in N; C/D: VGPR r holds M=r (lanes 0-15) / M=r+8
// (lanes 16-31), N=lane%16.
// ---------------------------------------------------------------------------
__global__ __launch_bounds__(256)
void sl_gemm(const float* __restrict__ x, const float* __restrict__ w,
             const float* __restrict__ bias, const int* __restrict__ perm,
             const int4* __restrict__ tiles, float* __restrict__ out,
             int IN, int OUT) {
    __shared__ float As[2][BM][BK + KPAD];   // [buf][row][k]  18.4 KB
    __shared__ float Bs[2][BN][BK + KPAD];   // [buf][col][k]  36.9 KB (transposed W)
    __shared__ int rowIdx[BM];

    const int4 td  = tiles[blockIdx.x];
    const int subj = td.x, start = td.y, rowsValid = td.z;
    if (rowsValid == 0) return;
    const int n0 = blockIdx.y * BN;

    const int t     = threadIdx.x;
    const int lane  = t & 31;
    const int wave  = t >> 5;
    const int m2    = wave & 1;           // 0..1 : which 32-row half
    const int ncol  = wave >> 1;          // 0..3 : which 32-col slice
    const int l16   = lane & 15;
    const int khalf = (lane >> 4) << 1;   // 0 for lanes 0-15, 2 for lanes 16-31

    if (t < BM) {
        const int idx = (t < rowsValid) ? t : 0;   // clamp: masked at store
        rowIdx[t] = perm[start + idx];
    }
    __syncthreads();

    const float* __restrict__ wS = w + (size_t)subj * (size_t)IN * (size_t)OUT;

    // --- fixed staging coordinates for this thread ---
    const int am  = t >> 3;            // A rows am and am+32
    const int akq = (t & 7) << 2;      // A K offset 0,4,...,28
    const int kkB = t >> 5;            // B K rows kkB + 8r, r=0..3
    const int nq  = (lane) << 2;       // wait -- see below
    // B N offset: (t & 31) * 4
    const int nqB = (t & 31) << 2;

    // Gathered-row base pointers (fixed across K loop)
    const float* __restrict__ xp0 = x + (size_t)rowIdx[am]      * (size_t)IN + akq;
    const float* __restrict__ xp1 = x + (size_t)rowIdx[am + 32] * (size_t)IN + akq;
    const float* __restrict__ wp  = wS + (size_t)kkB * (size_t)OUT + n0 + nqB;
    const size_t wRowStep = (size_t)8 * (size_t)OUT;   // r stride (8 K rows)

    v8f acc00 = {}, acc01 = {}, acc10 = {}, acc11 = {};

    const int aRow = m2 * 32 + l16;       // fragment rows for this wave
    const int bCol = ncol * 32 + l16;     // fragment cols for this wave

    float4 aReg0, aReg1, bReg[4];

    // --- prologue: slab 0 -> registers -> LDS buffer 0 ---
    aReg0 = *(const float4*)(xp0);
    aReg1 = *(const float4*)(xp1);
    #pragma unroll
    for (int r = 0; r < 4; r++) bReg[r] = *(const float4*)(wp + (size_t)r * wRowStep);

    *(float4*)(&As[0][am][akq])      = aReg0;
    *(float4*)(&As[0][am + 32][akq]) = aReg1;
    #pragma unroll
    for (int r = 0; r < 4; r++) {
        const int kk = kkB + (r << 3);
        Bs[0][nqB + 0][kk] = bReg[r].x;
        Bs[0][nqB + 1][kk] = bReg[r].y;
        Bs[0][nqB + 2][kk] = bReg[r].z;
        Bs[0][nqB + 3][kk] = bReg[r].w;
    }
    __syncthreads();

    int p = 0;
    for (int k0 = 0; k0 < IN; k0 += BK) {
        const int kNext = k0 + BK;
        const bool hasNext = kNext < IN;

        // --- issue next slab's global loads (latency hidden by WMMAs) ---
        if (hasNext) {
            aReg0 = *(const float4*)(xp0 + kNext);
            aReg1 = *(const float4*)(xp1 + kNext);
            const float* wpk = wp + (size_t)kNext * (size_t)OUT;
            #pragma unroll
            for (int r = 0; r < 4; r++)
                bReg[r] = *(const float4*)(wpk + (size_t)r * wRowStep);
        }
        if (k0 + 2 * BK < IN)  // pull slab k0+2*BK toward GL2 (global_prefetch_b8)
            __builtin_prefetch(wp + (size_t)(k0 + 2 * BK) * (size_t)OUT, 0, 1);

        // --- compute current slab from LDS buffer p ---
        #pragma unroll
        for (int kk = 0; kk < BK; kk += 4) {
            const v2f a0 = *(const v2f*)(&As[p][aRow][kk + khalf]);
            const v2f a1 = *(const v2f*)(&As[p][aRow + 16][kk + khalf]);
            const v2f b0 = *(const v2f*)(&Bs[p][bCol][kk + khalf]);
            const v2f b1 = *(const v2f*)(&Bs[p][bCol + 16][kk + khalf]);
            acc00 = __builtin_amdgcn_wmma_f32_16x16x4_f32(
                false, a0, false, b0, (short)0, acc00, false, false);
            acc01 = __builtin_amdgcn_wmma_f32_16x16x4_f32(
                false, a0, false, b1, (short)0, acc01, false, false);
            acc10 = __builtin_amdgcn_wmma_f32_16x16x4_f32(
                false, a1, false, b0, (short)0, acc10, false, false);
            acc11 = __builtin_amdgcn_wmma_f32_16x16x4_f32(
                false, a1, false, b1, (short)0, acc11, false, false);
        }

        // --- drain staged registers into the other LDS buffer ---
        if (hasNext) {
            const int q = p ^ 1;
            *(float4*)(&As[q][am][akq])      = aReg0;
            *(float4*)(&As[q][am + 32][akq]) = aReg1;
            #pragma unroll
            for (int r = 0; r < 4; r++) {
                const int kk = kkB + (r << 3);
                Bs[q][nqB + 0][kk] = bReg[r].x;
                Bs[q][nqB + 1][kk] = bReg[r].y;
                Bs[q][nqB + 2][kk] = bReg[r].z;
                Bs[q][nqB + 3][kk] = bReg[r].w;
            }
        }
        __syncthreads();
        p ^= 1;
    }

    // --- epilogue: bias add + scatter via permutation ---
    const int colBase = n0 + (ncol << 5);
    const float* __restrict__ bS = bias + (size_t)subj * (size_t)OUT;
    const int c0 = colBase + l16;
    #pragma unroll
    for (int r = 0; r < 8; r++) {
        const int M = (lane < 16) ? r : (r + 8);
        // first 16-row subtile of this wave's half
        int rt = (m2 << 5) + M;
        if (rt < rowsValid) {
            const int gr = rowIdx[rt];
            out[(size_t)gr * (size_t)OUT + c0]      = acc00[r] + bS[c0];
            out[(size_t)gr * (size_t)OUT + c0 + 16] = acc01[r] + bS[c0 + 16];
        }
        // second 16-row subtile
        rt = (m2 << 5) + 16 + M;
        if (rt < rowsValid) {
            const int gr = rowIdx[rt];
            out[(size_t)gr * (size_t)OUT + c0]      = acc10[r] + bS[c0];
            out[(size_t)gr * (size_t)OUT + c0 + 16] = acc11[r] + bS[c0 + 16];
        }
    }
}

// ---------------------------------------------------------------------------
extern "C" void kernel_launch(void* const* d_in, const int* in_sizes, int n_in,
                              void* d_out, int out_size, void* d_ws, size_t ws_size,
                              hipStream_t stream) {
    const float* x    = (const float*)d_in[0];
    const int*   sid  = (const int*)d_in[1];     // subject ids (integer input)
    const float* w    = (const float*)d_in[2];   // [S, IN, OUT]
    const float* bias = (const float*)d_in[3];   // [S, OUT]
    float* out = (float*)d_out;

    const int B   = in_sizes[1];              // 8192
    const int IN  = in_sizes[0] / B;          // 2048
    const int S   = NSUBJ;                    // 8
    const int OUT = in_sizes[3] / S;          // 3072

    // Workspace: perm[B] ints, then tile descriptors (int4), 16B aligned.
    int*  perm  = (int*)d_ws;
    const int maxT = B / BM + S;              // 136
    int4* tiles = (int4*)((char*)d_ws + (size_t)B * sizeof(int));

    sl_build_perm<<<1, 256, 0, stream>>>(sid, perm, tiles, B, maxT);

    dim3 grid(maxT, OUT / BN);                // 136 x 24
    sl_gemm<<<grid, 256, 0, stream>>>(x, w, bias, perm, tiles, out, IN, OUT);
}